// SRNN_Softmax_10041633538113
// MI455X (gfx1250) — compile-verified
//
#include <hip/hip_runtime.h>
#include <hip/hip_bf16.h>
#include <math.h>

typedef float v2f __attribute__((ext_vector_type(2)));
typedef float v8f __attribute__((ext_vector_type(8)));

constexpr int HN = 4096;   // H == V == O
constexpr int MS = 104;    // stack depth

// Zero-pad sizes: one full per-wave K-slice + ds_load reach (max k + 7)
constexpr int ZPAD_H = 520;   // kernel 1: K-slice 512
constexpr int ZPAD_Y = 264;   // kernel 2: K-slice 256

// ---------------------------------------------------------------------------
// Kernel 1: h = tanh(W_ih@emb + b_ih + W_hh@hbar + b_hh)
//   hbar = W_sh*stack[0] + b_sh + hidden0 (built on the fly into LDS)
//   grid = 256 row-blocks (16 rows each), 512 threads = 16 waves.
//   Waves 0..7 stream W_ih (K slice 512 each), waves 8..15 stream W_hh.
//   A tiles loaded directly in V_WMMA_F32_16X16X4_F32 lane layout (b64/lane),
//   vector in B column 0 (lanes 0,16), other lanes read a zeroed LDS pad.
// ---------------------------------------------------------------------------
__global__ __launch_bounds__(512)
void gemv_h_kernel(const int* __restrict__ inp,
                   const float* __restrict__ hidden0,
                   const float* __restrict__ stack,
                   const float* __restrict__ E,
                   const float* __restrict__ W_ih,
                   const float* __restrict__ b_ih,
                   const float* __restrict__ W_hh,
                   const float* __restrict__ b_hh,
                   const float* __restrict__ W_sh,
                   const float* __restrict__ b_sh,
                   float* __restrict__ h_out)
{
    __shared__ float xcat[2 * HN + ZPAD_H];  // [emb ; hbar ; zeros]
    __shared__ float red[16 * 16];           // per-wave partials

    const int tid  = threadIdx.x;
    const int lane = tid & 31;
    const int wid  = tid >> 5;

    // Stage emb row, hbar, and the zero pad into LDS
    const float* erow = E + (size_t)inp[0] * HN;
    const float  s0   = stack[0];
    for (int j = tid; j < HN; j += 512) {
        xcat[j]      = erow[j];
        xcat[HN + j] = W_sh[j] * s0 + b_sh[j] + hidden0[j];
    }
    for (int j = tid; j < ZPAD_H; j += 512) xcat[2 * HN + j] = 0.f;
    __syncthreads();

    const int rowBase = blockIdx.x * 16;
    const int r       = rowBase + (lane & 15);
    const float* Wmat = (wid < 8) ? W_ih : W_hh;
    const int kbase   = (wid & 7) * 512;                 // K slice in this matrix
    const int xslice  = ((wid < 8) ? 0 : HN) + kbase;    // slice in concatenated x
    const bool hi   = lane >= 16;
    const bool col0 = (lane & 15) == 0;

    // A base: lanes<16 supply (k,k+1), lanes>=16 supply (k+2,k+3) of row r
    const float* wrow = Wmat + (size_t)r * HN + kbase + ((lane >> 4) << 1);
    // B base: col-0 lanes read the x slice, all others read the zero pad
    const int xb = (col0 ? xslice : 2 * HN) + ((lane >> 4) << 1);

    v8f c0 = {};   // accumulators; only column 0 (lanes 0,16) is live
    v8f c1 = {};

    #pragma unroll 8
    for (int k = 0; k < 512; k += 8) {
        v2f A0 = *(const v2f*)(wrow + k);        // K = kb..kb+3 tile (global_load_b64)
        v2f A1 = *(const v2f*)(wrow + k + 4);    // K = kb+4..kb+7 tile
        v2f B0, B1;
        B0.x = xcat[xb + k];
        B0.y = xcat[xb + k + 1];
        B1.x = xcat[xb + k + 4];
        B1.y = xcat[xb + k + 5];
        c0 = __builtin_amdgcn_wmma_f32_16x16x4_f32(false, A0, false, B0,
                                                   (short)0, c0, false, false);
        c1 = __builtin_amdgcn_wmma_f32_16x16x4_f32(false, A1, false, B1,
                                                   (short)0, c1, false, false);
    }

    // column 0 of C: lane0 holds rows M=0..7, lane16 holds rows M=8..15
    if (col0) {
        const int mb = hi ? 8 : 0;
        #pragma unroll
        for (int j = 0; j < 8; ++j) red[wid * 16 + mb + j] = c0[j] + c1[j];
    }
    __syncthreads();

    if (tid < 16) {
        float y = 0.f;
        #pragma unroll
        for (int w = 0; w < 16; ++w) y += red[w * 16 + tid];
        const int row = rowBase + tid;
        h_out[row] = tanhf(y + b_ih[row] + b_hh[row]);
    }
}

// ---------------------------------------------------------------------------
// Kernel 2: output = sigmoid(W_y @ h + b_y)   (same WMMA-GEMV scheme)
// ---------------------------------------------------------------------------
__global__ __launch_bounds__(512)
void gemv_y_kernel(const float* __restrict__ W_y,
                   const float* __restrict__ b_y,
                   const float* __restrict__ h_in,
                   float* __restrict__ out)
{
    __shared__ float xv[HN + ZPAD_Y];
    __shared__ float red[16 * 16];

    const int tid  = threadIdx.x;
    const int lane = tid & 31;
    const int wid  = tid >> 5;

    for (int j = tid; j < HN; j += 512) xv[j] = h_in[j];
    for (int j = tid; j < ZPAD_Y; j += 512) xv[HN + j] = 0.f;
    __syncthreads();

    const int rowBase = blockIdx.x * 16;
    const int r       = rowBase + (lane & 15);
    const int kbase   = wid * 256;
    const bool hi   = lane >= 16;
    const bool col0 = (lane & 15) == 0;

    const float* wrow = W_y + (size_t)r * HN + kbase + ((lane >> 4) << 1);
    const int xb = (col0 ? kbase : HN) + ((lane >> 4) << 1);

    v8f c0 = {};
    v8f c1 = {};

    #pragma unroll 8
    for (int k = 0; k < 256; k += 8) {
        v2f A0 = *(const v2f*)(wrow + k);
        v2f A1 = *(const v2f*)(wrow + k + 4);
        v2f B0, B1;
        B0.x = xv[xb + k];
        B0.y = xv[xb + k + 1];
        B1.x = xv[xb + k + 4];
        B1.y = xv[xb + k + 5];
        c0 = __builtin_amdgcn_wmma_f32_16x16x4_f32(false, A0, false, B0,
                                                   (short)0, c0, false, false);
        c1 = __builtin_amdgcn_wmma_f32_16x16x4_f32(false, A1, false, B1,
                                                   (short)0, c1, false, false);
    }

    if (col0) {
        const int mb = hi ? 8 : 0;
        #pragma unroll
        for (int j = 0; j < 8; ++j) red[wid * 16 + mb + j] = c0[j] + c1[j];
    }
    __syncthreads();

    if (tid < 16) {
        float y = 0.f;
        #pragma unroll
        for (int w = 0; w < 16; ++w) y += red[w * 16 + tid];
        const int row = rowBase + tid;
        const float z = y + b_y[row];
        out[row] = 1.f / (1.f + expf(-z));
    }
}

// ---------------------------------------------------------------------------
// Kernel 3: attention softmax (2 logits), new_elt, stack blend, weights.
// ---------------------------------------------------------------------------
__global__ __launch_bounds__(256)
void head_kernel(const float* __restrict__ h,
                 const float* __restrict__ W_n,
                 const float* __restrict__ b_n,
                 const float* __restrict__ W_a,
                 const float* __restrict__ b_a,
                 const float* __restrict__ stack,
                 float* __restrict__ out)   // full d_out base
{
    __shared__ float ra0[256], ra1[256], rn[256];
    __shared__ float s_aw0, s_aw1, s_ne;

    const int t = threadIdx.x;
    float p0 = 0.f, p1 = 0.f, pn = 0.f;
    for (int j = t; j < HN; j += 256) {
        const float hv = h[j];
        p0 += W_a[j]      * hv;
        p1 += W_a[HN + j] * hv;
        pn += W_n[j]      * hv;
    }
    ra0[t] = p0; ra1[t] = p1; rn[t] = pn;
    __syncthreads();
    for (int s = 128; s > 0; s >>= 1) {
        if (t < s) { ra0[t] += ra0[t+s]; ra1[t] += ra1[t+s]; rn[t] += rn[t+s]; }
        __syncthreads();
    }
    if (t == 0) {
        const float a0 = ra0[0] + b_a[0];
        const float a1 = ra1[0] + b_a[1];
        const float m  = fmaxf(a0, a1);
        const float e0 = expf(a0 - m), e1 = expf(a1 - m);
        const float inv = 1.f / (e0 + e1);
        s_aw0 = e0 * inv;
        s_aw1 = e1 * inv;
        s_ne  = 1.f / (1.f + expf(-(rn[0] + b_n[0])));
        out[2 * HN + MS] = s_aw0 + s_aw1;          // weights (index 8296)
    }
    __syncthreads();
    if (t < MS) {
        const float push = (t == 0) ? s_ne : stack[t - 1];
        const float pop  = (t < MS - 1) ? stack[t + 1] : 0.f;
        out[2 * HN + t] = s_aw0 * push + s_aw1 * pop;   // new_stack
    }
}

// ---------------------------------------------------------------------------
extern "C" void kernel_launch(void* const* d_in, const int* in_sizes, int n_in,
                              void* d_out, int out_size, void* d_ws, size_t ws_size,
                              hipStream_t stream) {
    const int*   inp     = (const int*)  d_in[0];
    const float* hidden0 = (const float*)d_in[1];
    const float* stack   = (const float*)d_in[2];
    const float* E       = (const float*)d_in[3];
    const float* W_ih    = (const float*)d_in[4];
    const float* b_ih    = (const float*)d_in[5];
    const float* W_hh    = (const float*)d_in[6];
    const float* b_hh    = (const float*)d_in[7];
    const float* W_y     = (const float*)d_in[8];
    const float* b_y     = (const float*)d_in[9];
    const float* W_n     = (const float*)d_in[10];
    const float* b_n     = (const float*)d_in[11];
    const float* W_a     = (const float*)d_in[12];
    const float* b_a     = (const float*)d_in[13];
    const float* W_sh    = (const float*)d_in[14];
    const float* b_sh    = (const float*)d_in[15];

    float* out = (float*)d_out;
    float* h   = out + HN;   // hidden output slot doubles as h scratch

    gemv_h_kernel<<<HN / 16, 512, 0, stream>>>(inp, hidden0, stack, E,
                                               W_ih, b_ih, W_hh, b_hh,
                                               W_sh, b_sh, h);
    gemv_y_kernel<<<HN / 16, 512, 0, stream>>>(W_y, b_y, h, out);
    head_kernel<<<1, 256, 0, stream>>>(h, W_n, b_n, W_a, b_a, stack, out);
}